// RENRG_36275293782365
// MI455X (gfx1250) — compile-verified
//
#include <hip/hip_runtime.h>

// ---------------------------------------------------------------------------
// Problem constants (match reference)
// ---------------------------------------------------------------------------
#define N_C   256
#define M_C   512
#define NXI_C 2048
#define L_C   1024
#define D_C   (2*NXI_C + L_C)   // 5120
#define EPSF  0.001f

typedef float v2f __attribute__((ext_vector_type(2)));
typedef float v8f __attribute__((ext_vector_type(8)));
typedef int   v4i __attribute__((ext_vector_type(4)));

// ---------------------------------------------------------------------------
// CDNA5 async global->LDS path (ASYNCcnt-tracked), guarded so the file always
// compiles; falls back to synchronous staging if the builtins are absent.
// Parameter types taken from the compiler's own diagnostics:
//   b32 : (int addrspace(1)*, int addrspace(3)*, imm offset, cpol)
//   b128: (v4i addrspace(1)*, v4i addrspace(3)*, imm offset, cpol)
// ---------------------------------------------------------------------------
#if __has_builtin(__builtin_amdgcn_global_load_async_to_lds_b32) && \
    __has_builtin(__builtin_amdgcn_global_load_async_to_lds_b128) && \
    __has_builtin(__builtin_amdgcn_s_wait_asynccnt)
#define HAVE_ASYNC_LDS 1
#else
#define HAVE_ASYNC_LDS 0
#endif

#if HAVE_ASYNC_LDS
typedef __attribute__((address_space(1))) int*  g32p;
typedef __attribute__((address_space(3))) int*  l32p;
typedef __attribute__((address_space(1))) v4i*  g128p;
typedef __attribute__((address_space(3))) v4i*  l128p;

__device__ __forceinline__ void async_cp_b32(const float* g, float* l) {
    __builtin_amdgcn_global_load_async_to_lds_b32((g32p)g, (l32p)l, 0, 0);
}
__device__ __forceinline__ void async_cp_b128(const float* g, float* l) {
    __builtin_amdgcn_global_load_async_to_lds_b128((g128p)g, (l128p)l, 0, 0);
}
#endif

// ---------------------------------------------------------------------------
// Generic fp32 WMMA GEMM:  C[M x N] (+)= alpha * op(A) @ op(B)
//   op(A)[m][k] = TA ? A[(aro+k)*lda + aco+m] : A[(aro+m)*lda + aco+k]
//   op(B)[k][n] = TB ? B[(bro+n)*ldb + bco+k] : B[(bro+k)*ldb + bco+n]
//   alpha = aconst * g^gpow   (g read from gptr; gptr may be null -> g=1)
// Requires M % 128 == 0, N % 64 == 0, K % 16 == 0 (all call sites satisfy).
// Block: 256 threads (8 waves). Block tile 128x64. Wave tile 32x32 (2x2 WMMA).
// Double-buffered LDS; staging via GLOBAL_LOAD_ASYNC_TO_LDS when available.
// ---------------------------------------------------------------------------
template<int TA, int TB>
__global__ __launch_bounds__(256) void gemm_f32_wmma(
    float* __restrict__ C, int ldc,
    const float* __restrict__ A, int lda, int aro, int aco,
    const float* __restrict__ B, int ldb, int bro, int bco,
    int K, float aconst, int gpow, const float* __restrict__ gptr, int accum)
{
    // Row pitches 132/68 floats = 528/272 bytes: multiples of 16 (B128-safe).
    __shared__ float As[2][16][128 + 4];
    __shared__ float Bs[2][16][64 + 4];

    const int tid  = threadIdx.x;
    const int lane = tid & 31;
    const int wave = tid >> 5;
    const int wm   = wave & 3;      // 0..3 -> row strip wm*32
    const int wn   = wave >> 2;    // 0..1 -> col strip wn*32
    const int bm0  = blockIdx.y * 128;
    const int bn0  = blockIdx.x * 64;

    // async instructions issued per stage (compile-time for s_wait immediates)
    constexpr int NIA = TA ? 2 : 8;   // A: 2x B128 (contig) or 8x B32 (scatter)
    constexpr int NIB = TB ? 4 : 1;   // B: 4x B32 (scatter) or 1x B128 (contig)
    constexpr int NI  = NIA + NIB;
    (void)NI;

    // ---- stage one 16-deep K slice into LDS buffer `buf` ----
    auto stage = [&](int buf, int k0) {
#if HAVE_ASYNC_LDS
        if (TA) {
            // global contiguous over m -> B128, 4 floats per lane, 2 issues
            #pragma unroll
            for (int i = 0; i < 2; ++i) {
                int idx4 = (i * 256 + tid) * 4;
                int k = idx4 >> 7, m = idx4 & 127;
                async_cp_b128(&A[(size_t)(aro + k0 + k) * lda + (aco + bm0 + m)],
                              &As[buf][k][m]);
            }
        } else {
            // transpose on the fly: per-lane independent global/LDS addresses
            #pragma unroll
            for (int i = 0; i < 8; ++i) {
                int idx = i * 256 + tid;
                int m = idx >> 4, k = idx & 15;       // coalesce global over k
                async_cp_b32(&A[(size_t)(aro + bm0 + m) * lda + (aco + k0 + k)],
                             &As[buf][k][m]);
            }
        }
        if (TB) {
            #pragma unroll
            for (int i = 0; i < 4; ++i) {
                int idx = i * 256 + tid;
                int n = idx >> 4, k = idx & 15;       // coalesce global over k
                async_cp_b32(&B[(size_t)(bro + bn0 + n) * ldb + (bco + k0 + k)],
                             &Bs[buf][k][n]);
            }
        } else {
            int idx4 = tid * 4;
            int k = idx4 >> 6, n = idx4 & 63;          // contiguous: one B128
            async_cp_b128(&B[(size_t)(bro + k0 + k) * ldb + (bco + bn0 + n)],
                          &Bs[buf][k][n]);
        }
#else
        if (TA) {
            for (int idx = tid; idx < 16 * 128; idx += 256) {
                int k = idx >> 7, m = idx & 127;
                As[buf][k][m] = A[(size_t)(aro + k0 + k) * lda + (aco + bm0 + m)];
            }
        } else {
            for (int idx = tid; idx < 16 * 128; idx += 256) {
                int m = idx >> 4, k = idx & 15;
                As[buf][k][m] = A[(size_t)(aro + bm0 + m) * lda + (aco + k0 + k)];
            }
        }
        if (TB) {
            for (int idx = tid; idx < 16 * 64; idx += 256) {
                int n = idx >> 4, k = idx & 15;
                Bs[buf][k][n] = B[(size_t)(bro + bn0 + n) * ldb + (bco + k0 + k)];
            }
        } else {
            for (int idx = tid; idx < 16 * 64; idx += 256) {
                int k = idx >> 6, n = idx & 63;
                Bs[buf][k][n] = B[(size_t)(bro + k0 + k) * ldb + (bco + bn0 + n)];
            }
        }
#endif
    };

    v8f acc00 = {}, acc01 = {}, acc10 = {}, acc11 = {};

    const int am = wm * 32 + (lane & 15);
    const int bn = wn * 32 + (lane & 15);

    stage(0, 0);
    for (int k0 = 0; k0 < K; k0 += 16) {
        const int buf = (k0 >> 4) & 1;
        const int nxt = k0 + 16;
        if (nxt < K) stage(buf ^ 1, nxt);
#if HAVE_ASYNC_LDS
        // Async ops complete in order: <= NI outstanding means the *current*
        // stage's transfers have landed in LDS (next stage's NI may remain).
        if (nxt < K) __builtin_amdgcn_s_wait_asynccnt(NI);
        else         __builtin_amdgcn_s_wait_asynccnt(0);
#endif
        __syncthreads();   // all waves' stage-k transfers visible to everyone

#pragma unroll
        for (int ks = 0; ks < 4; ++ks) {
            const int k2 = ks * 4 + ((lane >> 4) << 1);    // ISA fp32 frag K split
            v2f a0, a1, b0, b1;
            a0.x = As[buf][k2][am];        a0.y = As[buf][k2 + 1][am];
            a1.x = As[buf][k2][am + 16];   a1.y = As[buf][k2 + 1][am + 16];
            b0.x = Bs[buf][k2][bn];        b0.y = Bs[buf][k2 + 1][bn];
            b1.x = Bs[buf][k2][bn + 16];   b1.y = Bs[buf][k2 + 1][bn + 16];
            acc00 = __builtin_amdgcn_wmma_f32_16x16x4_f32(false, a0, false, b0, (short)0, acc00, false, false);
            acc01 = __builtin_amdgcn_wmma_f32_16x16x4_f32(false, a0, false, b1, (short)0, acc01, false, false);
            acc10 = __builtin_amdgcn_wmma_f32_16x16x4_f32(false, a1, false, b0, (short)0, acc10, false, false);
            acc11 = __builtin_amdgcn_wmma_f32_16x16x4_f32(false, a1, false, b1, (short)0, acc11, false, false);
        }
        __syncthreads();   // stage k fully consumed before its buffer is reused
    }

    const float g     = gptr ? gptr[0] : 1.0f;
    const float alpha = aconst * (gpow == 0 ? 1.0f : (gpow > 0 ? g : 1.0f / g));
    const int colBase = bn0 + wn * 32 + (lane & 15);
    const int rowHalf = (lane >> 4) * 8;                   // ISA C/D layout

#pragma unroll
    for (int tm = 0; tm < 2; ++tm) {
#pragma unroll
        for (int tn = 0; tn < 2; ++tn) {
            v8f av = (tm == 0) ? (tn == 0 ? acc00 : acc01)
                               : (tn == 0 ? acc10 : acc11);
            int col = colBase + tn * 16;
#pragma unroll
            for (int r = 0; r < 8; ++r) {
                int row = bm0 + wm * 32 + tm * 16 + rowHalf + r;
                size_t ci = (size_t)row * ldc + col;
                float v = alpha * av[r];
                C[ci] = accum ? (C[ci] + v) : v;
            }
        }
    }
}

// ---------------------------------------------------------------------------
// Elementwise helpers
// ---------------------------------------------------------------------------
__global__ void addY3eps_kernel(float* __restrict__ M, const float* __restrict__ Y3)
{
    int idx = blockIdx.x * 256 + threadIdx.x;
    if (idx >= N_C * N_C) return;
    int i = idx >> 8, j = idx & 255;
    M[idx] += Y3[idx] - Y3[j * N_C + i] + (i == j ? EPSF : 0.0f);
}

__global__ void imm_kernel(float* __restrict__ T, const float* __restrict__ M)
{   // T = I - M (256x256)
    int idx = blockIdx.x * 256 + threadIdx.x;
    if (idx >= N_C * N_C) return;
    int i = idx >> 8, j = idx & 255;
    T[idx] = (i == j ? 1.0f : 0.0f) - M[idx];
}

__global__ void build_aug_kernel(float* __restrict__ G, const float* __restrict__ A,
                                 float dconst, int dgpow, const float* __restrict__ gptr)
{   // G[256x512] = [A + dadd*I | I]
    int idx = blockIdx.x * 256 + threadIdx.x;
    if (idx >= 256 * 512) return;
    int i = idx >> 9, j = idx & 511;
    float g = gptr ? gptr[0] : 1.0f;
    float dadd = dconst * (dgpow == 0 ? 1.0f : (dgpow > 0 ? g : 1.0f / g));
    float v;
    if (j < 256) v = A[i * 256 + j] + (i == j ? dadd : 0.0f);
    else         v = ((j - 256) == i) ? 1.0f : 0.0f;
    G[idx] = v;
}

__global__ void scale_g_kernel(float* __restrict__ p, int n, const float* __restrict__ gptr)
{
    int idx = blockIdx.x * 256 + threadIdx.x;
    if (idx < n) p[idx] *= gptr[0];
}

__global__ void sub_inplace_kernel(float* __restrict__ a, const float* __restrict__ b, int n)
{
    int idx = blockIdx.x * 256 + threadIdx.x;
    if (idx < n) a[idx] -= b[idx];
}

__global__ void build_esys_kernel(float* __restrict__ Es, const float* __restrict__ EH,
                                  const float* __restrict__ Y, const float* __restrict__ exi)
{   // Es[2048 x 2049] = [ 0.5*(EH + Y - Y^T) + EPS*I | E_xi ]
    int idx = blockIdx.x * 256 + threadIdx.x;
    const int total = NXI_C * (NXI_C + 1);
    if (idx >= total) return;
    int i = idx / (NXI_C + 1);
    int j = idx - i * (NXI_C + 1);
    float v;
    if (j < NXI_C) {
        v = 0.5f * (EH[(size_t)i * NXI_C + j] + Y[(size_t)i * NXI_C + j] - Y[(size_t)j * NXI_C + i]);
        if (i == j) v += EPSF;
    } else {
        v = exi[i];
    }
    Es[idx] = v;
}

// ---------------------------------------------------------------------------
// 256x256 Gauss-Jordan inverse on augmented [A|I] (G is 256x512), 1 block/256 thr
// ---------------------------------------------------------------------------
__global__ __launch_bounds__(256) void gj_inverse_kernel(float* __restrict__ G)
{
    const int n = 256, w = 512;
    __shared__ float rowk[512];
    __shared__ float spiv;
    int tid = threadIdx.x;
    for (int k = 0; k < n; ++k) {
        if (tid == 0) spiv = G[(size_t)k * w + k];
        __syncthreads();
        float pinv = 1.0f / spiv;
        for (int j = tid; j < w; j += 256) {
            float v = G[(size_t)k * w + j] * pinv;
            rowk[j] = v;
            G[(size_t)k * w + j] = v;
        }
        __syncthreads();
        if (tid != k) {
            float f = G[(size_t)tid * w + k];
            for (int j = 0; j < w; ++j)
                G[(size_t)tid * w + j] -= f * rowk[j];
        }
        __syncthreads();
    }
}

// ---------------------------------------------------------------------------
// Row-dot matvec kernels (one block of 256 per output row)
// ---------------------------------------------------------------------------
__global__ __launch_bounds__(256) void c_kernel(float* __restrict__ c,
    const float* __restrict__ H21, const float* __restrict__ xi,
    const float* __restrict__ D12, const float* __restrict__ w)
{   // c[i] = -(H21 @ xi)[i] + (D12 @ w)[i]
    __shared__ float red[256];
    int i = blockIdx.x, tid = threadIdx.x;
    float s = 0.0f;
    for (int j = tid; j < NXI_C; j += 256) s -= H21[(size_t)i * NXI_C + j] * xi[j];
    s += D12[(size_t)i * N_C + tid] * w[tid];
    red[tid] = s; __syncthreads();
    for (int st = 128; st > 0; st >>= 1) { if (tid < st) red[tid] += red[tid + st]; __syncthreads(); }
    if (tid == 0) c[i] = red[0];
}

__global__ __launch_bounds__(256) void exi_kernel(float* __restrict__ exi,
    const float* __restrict__ Fm, const float* __restrict__ xi,
    const float* __restrict__ B1, const float* __restrict__ eps,
    const float* __restrict__ B2, const float* __restrict__ w)
{   // E_xi[i] = Fm@xi + B1@eps + B2@w
    __shared__ float red[256];
    int i = blockIdx.x, tid = threadIdx.x;
    float s = 0.0f;
    for (int j = tid; j < NXI_C; j += 256) s += Fm[(size_t)i * NXI_C + j] * xi[j];
    for (int j = tid; j < L_C;   j += 256) s += B1[(size_t)i * L_C + j] * eps[j];
    s += B2[(size_t)i * N_C + tid] * w[tid];
    red[tid] = s; __syncthreads();
    for (int st = 128; st > 0; st >>= 1) { if (tid < st) red[tid] += red[tid + st]; __syncthreads(); }
    if (tid == 0) exi[i] = red[0];
}

__global__ __launch_bounds__(256) void u_kernel(float* __restrict__ u,
    const float* __restrict__ C2, const float* __restrict__ xi,
    const float* __restrict__ D21, const float* __restrict__ eps,
    const float* __restrict__ D22, const float* __restrict__ w)
{   // u[i] = C2@xi + D21@eps + D22@w  (written to d_out[0:512])
    __shared__ float red[256];
    int i = blockIdx.x, tid = threadIdx.x;
    float s = 0.0f;
    for (int j = tid; j < NXI_C; j += 256) s += C2[(size_t)i * NXI_C + j] * xi[j];
    for (int j = tid; j < L_C;   j += 256) s += D21[(size_t)i * L_C + j] * eps[j];
    s += D22[(size_t)i * N_C + tid] * w[tid];
    red[tid] = s; __syncthreads();
    for (int st = 128; st > 0; st >>= 1) { if (tid < st) red[tid] += red[tid + st]; __syncthreads(); }
    if (tid == 0) u[i] = red[0];
}

// ---------------------------------------------------------------------------
// Sequential ReLU forward substitution: eps[i] = relu((c[i] + D11[i]@eps)/lam[i])
// D11 = -strict_lower(H22); lam[i] = H22[i][i] + EPS. One block, 1024 threads.
// ---------------------------------------------------------------------------
__global__ __launch_bounds__(1024) void scan_eps_kernel(float* __restrict__ eps,
    const float* __restrict__ H22, const float* __restrict__ c)
{
    __shared__ float epsS[L_C];
    __shared__ float red[1024];
    int tid = threadIdx.x;
    epsS[tid] = 0.0f;
    __syncthreads();
    for (int i = 0; i < L_C; ++i) {
        float p = (tid < i) ? (-H22[(size_t)i * L_C + tid]) * epsS[tid] : 0.0f;
        red[tid] = p;
        __syncthreads();
        for (int s = 512; s > 0; s >>= 1) {
            if (tid < s) red[tid] += red[tid + s];
            __syncthreads();
        }
        if (tid == 0) {
            float lam = H22[(size_t)i * L_C + i] + EPSF;
            float v = (c[i] + red[0]) / lam;
            epsS[i] = v > 0.0f ? v : 0.0f;
        }
        __syncthreads();
    }
    eps[tid] = epsS[tid];
}

// ---------------------------------------------------------------------------
// Solve E x = b via Gaussian elimination (no pivot) on augmented Es[2048x2049].
// One block, 1024 threads. x written to output.
// ---------------------------------------------------------------------------
__global__ __launch_bounds__(1024) void solve_kernel(float* __restrict__ Es, float* __restrict__ x)
{
    const int n = NXI_C, ld = NXI_C + 1;
    __shared__ float rowk[NXI_C + 1];
    __shared__ float xk;
    int tid = threadIdx.x;
    // forward elimination
    for (int k = 0; k < n; ++k) {
        for (int j = tid; j < ld; j += 1024) rowk[j] = Es[(size_t)k * ld + j];
        __syncthreads();
        float pinv = 1.0f / rowk[k];
        for (int r = k + 1 + tid; r < n; r += 1024) {
            size_t base = (size_t)r * ld;
            float f = Es[base + k] * pinv;
            for (int j = k; j < ld; ++j) Es[base + j] -= f * rowk[j];
        }
        __syncthreads();
    }
    // back substitution
    for (int k = n - 1; k >= 0; --k) {
        if (tid == 0) {
            float v = Es[(size_t)k * ld + n] / Es[(size_t)k * ld + k];
            x[k] = v; xk = v;
        }
        __syncthreads();
        float v = xk;
        for (int r = tid; r < k; r += 1024)
            Es[(size_t)r * ld + n] -= Es[(size_t)r * ld + k] * v;
        __syncthreads();
    }
}

// ---------------------------------------------------------------------------
// Host-side orchestration
// ---------------------------------------------------------------------------
static inline void launch_gemm(hipStream_t s, float* C, int ldc,
    const float* A, int lda, int aro, int aco, int tA,
    const float* B, int ldb, int bro, int bco, int tB,
    int M, int N, int K, float aconst, int gpow, const float* gptr, int accum)
{
    dim3 grid(N / 64, M / 128);
    if (tA) {
        if (tB) gemm_f32_wmma<1,1><<<grid, 256, 0, s>>>(C, ldc, A, lda, aro, aco, B, ldb, bro, bco, K, aconst, gpow, gptr, accum);
        else    gemm_f32_wmma<1,0><<<grid, 256, 0, s>>>(C, ldc, A, lda, aro, aco, B, ldb, bro, bco, K, aconst, gpow, gptr, accum);
    } else {
        if (tB) gemm_f32_wmma<0,1><<<grid, 256, 0, s>>>(C, ldc, A, lda, aro, aco, B, ldb, bro, bco, K, aconst, gpow, gptr, accum);
        else    gemm_f32_wmma<0,0><<<grid, 256, 0, s>>>(C, ldc, A, lda, aro, aco, B, ldb, bro, bco, K, aconst, gpow, gptr, accum);
    }
}

extern "C" void kernel_launch(void* const* d_in, const int* in_sizes, int n_in,
                              void* d_out, int out_size, void* d_ws, size_t ws_size,
                              hipStream_t stream)
{
    (void)in_sizes; (void)n_in; (void)out_size; (void)ws_size;

    const float* w   = (const float*)d_in[0];   // (256,)
    const float* xi  = (const float*)d_in[1];   // (2048,)
    const float* gp  = (const float*)d_in[2];   // (1,)
    const float* X   = (const float*)d_in[3];   // (5120,5120)
    const float* Y   = (const float*)d_in[4];   // (2048,2048)
    const float* B2  = (const float*)d_in[5];   // (2048,256)
    const float* C2  = (const float*)d_in[6];   // (512,2048)
    const float* D21 = (const float*)d_in[7];   // (512,1024)
    const float* Z3  = (const float*)d_in[8];   // (256,256)
    const float* X3  = (const float*)d_in[9];   // (256,256)
    const float* Y3  = (const float*)d_in[10];  // (256,256)
    const float* D12 = (const float*)d_in[11];  // (1024,256)

    float* out = (float*)d_out;                 // [u(512) | xi_(2048)]

    // ---- workspace layout (floats) ----
    float* ws = (float*)d_ws;
    size_t o = 0;
    float* fMm  = ws + o; o += 256 * 256;
    float* fG1  = ws + o; o += 256 * 512;          // GJ for (I+M); Minv at col 256, ld 512
    float* fT1  = ws + o; o += 256 * 256;          // I - M
    float* fD22 = ws + o; o += 512 * 256;          // Nmat -> D22 (scaled by g)
    float* fRc  = ws + o; o += 256 * 256;
    float* fG2  = ws + o; o += 256 * 512;          // GJ for R_cap; Rinv at col 256
    float* fR1  = ws + o; o += 2048 * 256;         // vecR rows 0:2048
    float* fR2  = ws + o; o += 1024 * 256;         // vecR rows 2048:3072
    float* fW1a = ws + o; o += 2048 * 256;         // R1 @ Rinv
    float* fW1b = ws + o; o += 1024 * 256;         // R2 @ Rinv
    float* fW1c = ws + o; o += 2048 * 256;         // B2 @ Rinv
    float* fEH  = ws + o; o += (size_t)2048 * 2048;// H11 + H33
    float* fH21 = ws + o; o += (size_t)1024 * 2048;
    float* fH22 = ws + o; o += (size_t)1024 * 1024;
    float* fH31 = ws + o; o += (size_t)2048 * 2048;
    float* fH32 = ws + o; o += (size_t)2048 * 1024;
    float* fEs  = ws + o; o += (size_t)2048 * 2049;
    float* fc   = ws + o; o += 1024;
    float* fep  = ws + o; o += 1024;
    float* fexi = ws + o; o += 2048;

    // ---- 1) M = X3^T X3 + Z3^T Z3 + Y3 - Y3^T + EPS*I ----
    launch_gemm(stream, fMm, 256, X3, 256, 0, 0, 1, X3, 256, 0, 0, 0, 256, 256, 256, 1.0f, 0, nullptr, 0);
    launch_gemm(stream, fMm, 256, Z3, 256, 0, 0, 1, Z3, 256, 0, 0, 0, 256, 256, 256, 1.0f, 0, nullptr, 1);
    addY3eps_kernel<<<(256 * 256) / 256, 256, 0, stream>>>(fMm, Y3);

    // ---- 2) Minv = inv(I + M) via Gauss-Jordan ----
    build_aug_kernel<<<(256 * 512) / 256, 256, 0, stream>>>(fG1, fMm, 1.0f, 0, nullptr);
    gj_inverse_kernel<<<1, 256, 0, stream>>>(fG1);
    // Minv := fG1 with (bro=0, bco=256, ldb=512)

    // ---- 3) Nmat = [ (I-M)Minv ; -2 Z3 Minv ];  D22 = g * Nmat ----
    imm_kernel<<<(256 * 256) / 256, 256, 0, stream>>>(fT1, fMm);
    launch_gemm(stream, fD22, 256, fT1, 256, 0, 0, 0, fG1, 512, 0, 256, 0, 256, 256, 256, 1.0f, 0, nullptr, 0);
    launch_gemm(stream, fD22 + 256 * 256, 256, Z3, 256, 0, 0, 0, fG1, 512, 0, 256, 0, 256, 256, 256, -2.0f, 0, nullptr, 0);
    scale_g_kernel<<<(512 * 256) / 256, 256, 0, stream>>>(fD22, 512 * 256, gp);

    // ---- 4) R_cap = g*I - (1/g) D22^T D22;  Rinv = inv(R_cap) ----
    launch_gemm(stream, fRc, 256, fD22, 256, 0, 0, 1, fD22, 256, 0, 0, 0, 256, 256, 512, -1.0f, -1, gp, 0);
    build_aug_kernel<<<(256 * 512) / 256, 256, 0, stream>>>(fG2, fRc, 1.0f, +1, gp);
    gj_inverse_kernel<<<1, 256, 0, stream>>>(fG2);
    // Rinv := fG2 with (bro=0, bco=256, ldb=512)

    // ---- 5) vecR pieces: R1 = -(1/g) C2^T D22 ; R2 = -(1/g) D21^T D22 - D12 ; R3 = B2 ----
    launch_gemm(stream, fR1, 256, C2, 2048, 0, 0, 1, fD22, 256, 0, 0, 0, 2048, 256, 512, -1.0f, -1, gp, 0);
    launch_gemm(stream, fR2, 256, D21, 1024, 0, 0, 1, fD22, 256, 0, 0, 0, 1024, 256, 512, -1.0f, -1, gp, 0);
    sub_inplace_kernel<<<(1024 * 256) / 256, 256, 0, stream>>>(fR2, D12, 1024 * 256);

    // ---- 6) W1 = vecR @ Rinv pieces ----
    launch_gemm(stream, fW1a, 256, fR1, 256, 0, 0, 0, fG2, 512, 0, 256, 0, 2048, 256, 256, 1.0f, 0, nullptr, 0);
    launch_gemm(stream, fW1b, 256, fR2, 256, 0, 0, 0, fG2, 512, 0, 256, 0, 1024, 256, 256, 1.0f, 0, nullptr, 0);
    launch_gemm(stream, fW1c, 256, B2, 256, 0, 0, 0, fG2, 512, 0, 256, 0, 2048, 256, 256, 1.0f, 0, nullptr, 0);

    // ---- 7) H blocks: H = X^T X + W vecR^T + (1/g) vecQ vecQ^T (+EPS on diag) ----
    // EH = H11 + H33
    launch_gemm(stream, fEH, 2048, X, D_C, 0, 0, 1,    X, D_C, 0, 0, 0,    2048, 2048, D_C, 1.0f, 0, nullptr, 0);
    launch_gemm(stream, fEH, 2048, X, D_C, 0, 3072, 1, X, D_C, 0, 3072, 0, 2048, 2048, D_C, 1.0f, 0, nullptr, 1);
    launch_gemm(stream, fEH, 2048, fW1a, 256, 0, 0, 0, fR1, 256, 0, 0, 1,  2048, 2048, 256, 1.0f, 0, nullptr, 1);
    launch_gemm(stream, fEH, 2048, fW1c, 256, 0, 0, 0, B2, 256, 0, 0, 1,   2048, 2048, 256, 1.0f, 0, nullptr, 1);
    launch_gemm(stream, fEH, 2048, C2, 2048, 0, 0, 1,  C2, 2048, 0, 0, 0,  2048, 2048, 512, 1.0f, -1, gp, 1);
    // H21
    launch_gemm(stream, fH21, 2048, X, D_C, 0, 2048, 1, X, D_C, 0, 0, 0,   1024, 2048, D_C, 1.0f, 0, nullptr, 0);
    launch_gemm(stream, fH21, 2048, fW1b, 256, 0, 0, 0, fR1, 256, 0, 0, 1, 1024, 2048, 256, 1.0f, 0, nullptr, 1);
    launch_gemm(stream, fH21, 2048, D21, 1024, 0, 0, 1, C2, 2048, 0, 0, 0, 1024, 2048, 512, 1.0f, -1, gp, 1);
    // H22
    launch_gemm(stream, fH22, 1024, X, D_C, 0, 2048, 1, X, D_C, 0, 2048, 0, 1024, 1024, D_C, 1.0f, 0, nullptr, 0);
    launch_gemm(stream, fH22, 1024, fW1b, 256, 0, 0, 0, fR2, 256, 0, 0, 1,  1024, 1024, 256, 1.0f, 0, nullptr, 1);
    launch_gemm(stream, fH22, 1024, D21, 1024, 0, 0, 1, D21, 1024, 0, 0, 0, 1024, 1024, 512, 1.0f, -1, gp, 1);
    // H31 (= Fm)
    launch_gemm(stream, fH31, 2048, X, D_C, 0, 3072, 1, X, D_C, 0, 0, 0,   2048, 2048, D_C, 1.0f, 0, nullptr, 0);
    launch_gemm(stream, fH31, 2048, fW1c, 256, 0, 0, 0, fR1, 256, 0, 0, 1, 2048, 2048, 256, 1.0f, 0, nullptr, 1);
    // H32 (= B1)
    launch_gemm(stream, fH32, 1024, X, D_C, 0, 3072, 1, X, D_C, 0, 2048, 0, 2048, 1024, D_C, 1.0f, 0, nullptr, 0);
    launch_gemm(stream, fH32, 1024, fW1c, 256, 0, 0, 0, fR2, 256, 0, 0, 1,  2048, 1024, 256, 1.0f, 0, nullptr, 1);

    // ---- 8) c = -H21 @ xi + D12 @ w ; eps scan ----
    c_kernel<<<1024, 256, 0, stream>>>(fc, fH21, xi, D12, w);
    scan_eps_kernel<<<1, 1024, 0, stream>>>(fep, fH22, fc);

    // ---- 9) E_xi = Fm@xi + B1@eps + B2@w ; solve E xi_ = E_xi ----
    exi_kernel<<<2048, 256, 0, stream>>>(fexi, fH31, xi, fH32, fep, B2, w);
    build_esys_kernel<<<((size_t)2048 * 2049 + 255) / 256, 256, 0, stream>>>(fEs, fEH, Y, fexi);
    solve_kernel<<<1, 1024, 0, stream>>>(fEs, out + 512);   // xi_ -> out[512:2560]

    // ---- 10) u = C2@xi + D21@eps + D22@w -> out[0:512] ----
    u_kernel<<<512, 256, 0, stream>>>(out, C2, xi, D21, fep, fD22, w);
}